// Sage_11081015624123
// MI455X (gfx1250) — compile-verified
//
#include <hip/hip_runtime.h>
#include <hip/hip_bf16.h>
#include <stdint.h>

typedef float v2f __attribute__((ext_vector_type(2)));
typedef float v8f __attribute__((ext_vector_type(8)));

// ---------------------------------------------------------------------------
// Utility: zero a float buffer
// ---------------------------------------------------------------------------
__global__ void sage_zero_kernel(float* __restrict__ p, long n) {
    long i = (long)blockIdx.x * blockDim.x + threadIdx.x;
    if (i < n) p[i] = 0.0f;
}

// ---------------------------------------------------------------------------
// Degree: deg[dst[e]] += 1
// ---------------------------------------------------------------------------
__global__ void sage_degree_kernel(const long long* __restrict__ dst,
                                   float* __restrict__ deg, int E) {
    int e = blockIdx.x * blockDim.x + threadIdx.x;
    if (e < E) atomicAdd(&deg[(int)dst[e]], 1.0f);
}

// deg[i] = 1 / max(deg[i], 1)
__global__ void sage_invdeg_kernel(float* __restrict__ deg, int n) {
    int i = blockIdx.x * blockDim.x + threadIdx.x;
    if (i < n) deg[i] = 1.0f / fmaxf(deg[i], 1.0f);
}

// ---------------------------------------------------------------------------
// Scatter-sum: agg[dst[e], :] += feat[src[e], :]   (F == 128)
// One wave32 per edge; each lane: one b128 gather + 4 f32 atomics.
// ---------------------------------------------------------------------------
__global__ void sage_scatter_kernel(const float* __restrict__ feat,
                                    const long long* __restrict__ src,
                                    const long long* __restrict__ dst,
                                    float* __restrict__ agg, int E) {
    int wave = (blockIdx.x * blockDim.x + threadIdx.x) >> 5;
    int lane = threadIdx.x & 31;
    if (wave >= E) return;
    int s = (int)src[wave];
    int d = (int)dst[wave];
    const float4 v = *reinterpret_cast<const float4*>(feat + (long)s * 128 + lane * 4);
    float* ap = agg + (long)d * 128 + lane * 4;
    atomicAdd(ap + 0, v.x);
    atomicAdd(ap + 1, v.y);
    atomicAdd(ap + 2, v.z);
    atomicAdd(ap + 3, v.w);
}

// ---------------------------------------------------------------------------
// Scale rows: agg[i, :] *= inv[i]   (128 floats/row, one float4 per thread)
// ---------------------------------------------------------------------------
__global__ void sage_scale_kernel(float* __restrict__ agg,
                                  const float* __restrict__ inv, int n) {
    int idx  = blockIdx.x * blockDim.x + threadIdx.x;
    int node = idx >> 5;          // 32 float4s per row
    int c4   = idx & 31;
    if (node >= n) return;
    float s = inv[node];
    float4* p = reinterpret_cast<float4*>(agg + (long)node * 128) + c4;
    float4 v = *p;
    v.x *= s; v.y *= s; v.z *= s; v.w *= s;
    *p = v;
}

// ---------------------------------------------------------------------------
// Build combined transposed weights: Wt[nc][k] (k in [0,256)):
//   k <  128 : Wl[k][nc]
//   k >= 128 : Wr[k-128][nc]
// so the WMMA B-fragment is one aligned b64 load per k-step.
// ---------------------------------------------------------------------------
__global__ void sage_wt_kernel(const float* __restrict__ Wl,
                               const float* __restrict__ Wr,
                               float* __restrict__ Wt, int ncols) {
    int idx = blockIdx.x * blockDim.x + threadIdx.x;
    if (idx >= ncols * 256) return;
    int nc = idx >> 8;
    int k  = idx & 255;
    float v = (k < 128) ? Wl[(long)k * ncols + nc]
                        : Wr[(long)(k - 128) * ncols + nc];
    Wt[(long)nc * 256 + k] = v;
}

// ---------------------------------------------------------------------------
// Fused SAGE layer GEMM:
//   out = maybe_relu( [agg_scaled | self] (n x 256) @ Wt^T + bias )
// Block = 128 threads (4 waves) = one 16-row output stripe.
// Each wave owns 16-wide column tiles; 64 x v_wmma_f32_16x16x4_f32 per tile.
//
// A tile is staged with CDNA5 async global->LDS b128 copies (ASYNCcnt),
// fenced with s_wait_asynccnt before the workgroup barrier.
//
// WMMA f32 16x16x4 VGPR layouts (ISA 7.12.2):
//   A (16x4):  lanes 0-15: M=lane, VGPRs {K=k, K=k+1};
//              lanes 16-31: M=lane-16, VGPRs {K=k+2, K=k+3}
//   B (4x16):  mirrored K pairing, N = lane%16
//   C/D:       VGPR i <-> M = i + 8*(lane>=16), N = lane%16
// ---------------------------------------------------------------------------
__global__ __launch_bounds__(128)
void sage_wmma_kernel(const float* __restrict__ aggScaled, // [n,128]
                      const float* __restrict__ selfFeat,  // [n,128]
                      const float* __restrict__ Wt,        // [ncols,256]
                      const float* __restrict__ bias,      // [ncols]
                      float* __restrict__ out,             // [n,ncols]
                      int n, int ncols, int dorelu) {
    // 16 rows x 256 cols of A, rows padded to 260 dwords (260 % 64 = 4) so a
    // 16-row column read hits 16 distinct LDS banks. 260 % 4 == 0 keeps
    // b128 packets 16B-aligned and v2f reads 8B-aligned.
    __shared__ float At[16][260];

    const int mBase = blockIdx.x * 16;
    const int tid   = threadIdx.x;

    // LDS byte offset of At (sole shared object) via addrspace(3) cast.
    unsigned ldsBase = (unsigned)(uintptr_t)
        (__attribute__((address_space(3))) char*)&At[0][0];

    // Stage A = [agg_scaled | self]: 1024 async b128 packets, 8 per thread.
    for (int idx = tid; idx < 16 * 64; idx += 128) {
        int r  = idx >> 6;        // row 0..15
        int c4 = idx & 63;        // float4 column 0..63
        int g  = mBase + r;
        unsigned laddr = ldsBase + (unsigned)(r * 260 + c4 * 4) * 4u;
        if (g < n) {
            const float* gp = (c4 < 32)
                ? aggScaled + (long)g * 128 + c4 * 4
                : selfFeat  + (long)g * 128 + (c4 - 32) * 4;
            unsigned long long ga = (unsigned long long)(uintptr_t)gp;
            asm volatile("global_load_async_to_lds_b128 %0, %1, off"
                         :: "v"(laddr), "v"(ga) : "memory");
        } else {
            *reinterpret_cast<float4*>(&At[r][c4 * 4]) =
                make_float4(0.f, 0.f, 0.f, 0.f);
        }
    }
    asm volatile("s_wait_asynccnt 0x0" ::: "memory");
    __syncthreads();

    const int wave = tid >> 5;
    const int lane = tid & 31;
    const int half = lane >> 4;   // 0: lanes 0-15, 1: lanes 16-31
    const int lrow = lane & 15;   // M (for A) / N (for B,C,D) within tile
    const int ntiles = ncols >> 4;
    const bool fullStripe = (mBase + 16 <= n);

    for (int tn = wave; tn < ntiles; tn += 4) {
        const int nc = tn * 16 + lrow;
        const float* wrow = Wt + (long)nc * 256;
        v8f acc = {};
        #pragma unroll 8
        for (int k = 0; k < 256; k += 4) {
            const int k0 = k + half * 2;
            v2f a = *reinterpret_cast<const v2f*>(&At[lrow][k0]);   // ds b64
            v2f b = *reinterpret_cast<const v2f*>(wrow + k0);       // global b64
            acc = __builtin_amdgcn_wmma_f32_16x16x4_f32(
                false, a, false, b, (short)0, acc, false, false);
        }
        const float bv = bias[nc];
        float* obase = out + (long)(mBase + half * 8) * ncols + nc;
        if (fullStripe) {
            #pragma unroll
            for (int i = 0; i < 8; ++i) {
                float v = acc[i] + bv;
                if (dorelu) v = fmaxf(v, 0.0f);
                obase[(long)i * ncols] = v;
            }
        } else {
            #pragma unroll
            for (int i = 0; i < 8; ++i) {
                const int g = mBase + half * 8 + i;
                if (g < n) {
                    float v = acc[i] + bv;
                    if (dorelu) v = fmaxf(v, 0.0f);
                    out[(long)g * ncols + nc] = v;
                }
            }
        }
    }
}

// ---------------------------------------------------------------------------
// Launch
// ---------------------------------------------------------------------------
extern "C" void kernel_launch(void* const* d_in, const int* in_sizes, int n_in,
                              void* d_out, int out_size, void* d_ws, size_t ws_size,
                              hipStream_t stream) {
    const float*      x   = (const float*)d_in[0];
    const long long*  ei  = (const long long*)d_in[1];   // int64 [2, E]
    const float*      W1l = (const float*)d_in[2];
    const float*      b1  = (const float*)d_in[3];
    const float*      W1r = (const float*)d_in[4];
    const float*      W2l = (const float*)d_in[5];
    const float*      b2  = (const float*)d_in[6];
    const float*      W2r = (const float*)d_in[7];
    float*            out = (float*)d_out;

    const int F = 128;
    const int n = in_sizes[0] / F;       // 100000
    const int e = in_sizes[1] / 2;       // 600000

    const long long* src = ei;
    const long long* dst = ei + e;

    // Workspace layout (floats): [deg(npad) | agg(n*128) | h(n*128) | Wt(128*256)]
    float* ws  = (float*)d_ws;
    long  npad = ((long)n + 127) & ~127L;
    float* deg = ws;
    float* agg = ws + npad;
    float* h   = agg + (long)n * F;
    float* Wt  = h   + (long)n * F;

    const int Z = 256;

    // Zero deg + agg in one pass.
    {
        long tot = npad + (long)n * F;
        sage_zero_kernel<<<(int)((tot + Z - 1) / Z), Z, 0, stream>>>(ws, tot);
    }
    sage_degree_kernel<<<(e + Z - 1) / Z, Z, 0, stream>>>(dst, deg, e);
    sage_scatter_kernel<<<(int)(((long)e * 32 + Z - 1) / Z), Z, 0, stream>>>(x, src, dst, agg, e);
    sage_invdeg_kernel<<<(n + Z - 1) / Z, Z, 0, stream>>>(deg, n);
    sage_scale_kernel<<<(int)(((long)n * 32 + Z - 1) / Z), Z, 0, stream>>>(agg, deg, n);

    // Layer 1: h = relu(agg_scaled @ W1l + b1 + x @ W1r), ncols = 128
    sage_wt_kernel<<<(128 * 256 + Z - 1) / Z, Z, 0, stream>>>(W1l, W1r, Wt, 128);
    sage_wmma_kernel<<<(n + 15) / 16, 128, 0, stream>>>(agg, x, Wt, b1, h, n, 128, 1);

    // Re-zero agg, scatter h, scale.
    {
        long tot = (long)n * F;
        sage_zero_kernel<<<(int)((tot + Z - 1) / Z), Z, 0, stream>>>(agg, tot);
    }
    sage_scatter_kernel<<<(int)(((long)e * 32 + Z - 1) / Z), Z, 0, stream>>>(h, src, dst, agg, e);
    sage_scale_kernel<<<(int)(((long)n * 32 + Z - 1) / Z), Z, 0, stream>>>(agg, deg, n);

    // Layer 2: out = agg_scaled @ W2l + b2 + h @ W2r, ncols = 32, no relu
    sage_wt_kernel<<<(32 * 256 + Z - 1) / Z, Z, 0, stream>>>(W2l, W2r, Wt, 32);
    sage_wmma_kernel<<<(n + 15) / 16, 128, 0, stream>>>(agg, h, Wt, b2, out, n, 32, 0);
}